// LinearCRF_57535381897357
// MI455X (gfx1250) — compile-verified
//
#include <hip/hip_runtime.h>
#include <cstddef>

typedef float v2f __attribute__((ext_vector_type(2)));
typedef float v8f __attribute__((ext_vector_type(8)));

#define LBL    5
#define TSTART 3
#define TEND   4
#define LOG2E  1.44269504088896340736f
#define LN2    0.69314718055994530942f

// Raw hardware transcendentals (v_exp_f32 / v_log_f32 are base-2).
// Safe here: log inputs are either exactly 0 (-> -inf, desired) or in
// [~1e-2, ~2e2] (never subnormal); exp inputs are <= 0 and underflow-to-0
// is the desired semantics.
__device__ __forceinline__ float exp2r(float x) { return __builtin_amdgcn_exp2f(x); }
__device__ __forceinline__ float log2r(float x) { return __builtin_amdgcn_logf(x); }

// ---------------------------------------------------------------------------
// Forward (partition) kernel: one wave (32 threads) handles 16 sequences.
// Scaled-probability forward algorithm (alpha kept in base-2 units); per-step
// transition contraction done with V_WMMA_F32_16X16X4_F32:
//   D = E^T (16x4) x Aexp (4x16), batch on the N (lane) axis.
// A-matrix rows 8..15 duplicate rows 0..7, so D rows 8..15 duplicate the s[j]
// values into lanes 16..31: both wave halves run the identical recurrence and
// the B operand needs NO cross-lane shuffle (pure cndmask select).
// ---------------------------------------------------------------------------
__global__ void __launch_bounds__(32)
crf_forward_wmma(const float* __restrict__ lstm,
                 const float* __restrict__ trans,
                 const int*   __restrict__ lens,
                 float*       __restrict__ partials,
                 int B, int S)
{
  const int wave = blockIdx.x;
  const int lane = threadIdx.x;        // 0..31
  const int m    = lane & 15;          // batch slot within wave
  const int j    = lane & 7;           // A-matrix row within duplicated half
  int b = wave * 16 + m;
  const bool valid = (b < B) && (lane < 16);
  if (b >= B) b = B - 1;

  // Transition matrix in registers; only constant-indexed accesses below.
  float tr[LBL * LBL];
#pragma unroll
  for (int i = 0; i < LBL * LBL; ++i) tr[i] = trans[i];

  // A operand = E^T with row duplication: A[M][k] = exp(trans[src_k][M&7]),
  // src = {0,1,3,4} (PAD row contributes exactly 0).  16x4 f32 A layout:
  // lanes 0-15 hold M=lane with VGPR0=k0,VGPR1=k1; lanes 16-31 hold
  // M=lane-16 with VGPR0=k2,VGPR1=k3.
  const int r0 = (lane < 16) ? 0 : TSTART;   // k0 / k2 source labels
  const int r1 = (lane < 16) ? 1 : TEND;     // k1 / k3 source labels
  v2f Aop;
  Aop.x = (j < LBL) ? exp2r(trans[r0 * LBL + j] * LOG2E) : 0.f;
  Aop.y = (j < LBL) ? exp2r(trans[r1 * LBL + j] * LOG2E) : 0.f;

  const int len = lens[b];
  int gmax = len;                       // uniform loop bound for the wave
#pragma unroll
  for (int off = 16; off >= 1; off >>= 1)
    gmax = max(gmax, __shfl_xor(gmax, off, 32));

  // alpha0[j] = trans[START][j] + emit[b][0][j], stored in base-2 units.
  const float* ep = lstm + (size_t)b * S * LBL;
  float al0 = (tr[TSTART * LBL + 0] + ep[0]) * LOG2E;
  float al1 = (tr[TSTART * LBL + 1] + ep[1]) * LOG2E;
  float al3 = (tr[TSTART * LBL + 3] + ep[3]) * LOG2E;
  float al4 = (tr[TSTART * LBL + 4] + ep[4]) * LOG2E;

  // End-transition constants, pre-scaled to base-2 units.
  const float tE0 = tr[0 * LBL + TEND] * LOG2E;
  const float tE1 = tr[1 * LBL + TEND] * LOG2E;
  const float tE3 = tr[TSTART * LBL + TEND] * LOG2E;
  const float tE4 = tr[TEND * LBL + TEND] * LOG2E;

  float fin = 0.f;   // base-2 logsumexp of last_alpha + trans[:,END]
#define RECORD() do {                                                        \
    float x0 = al0 + tE0, x1 = al1 + tE1;                                    \
    float x3 = al3 + tE3, x4 = al4 + tE4;                                    \
    float mx2 = fmaxf(fmaxf(x0, x1), fmaxf(x3, x4));                         \
    fin = mx2 + log2r(exp2r(x0 - mx2) + exp2r(x1 - mx2) +                    \
                      exp2r(x3 - mx2) + exp2r(x4 - mx2));                    \
  } while (0)

  if (len == 1) RECORD();

  for (int t = 1; t < gmax; ++t) {
    // rescale: e[i] = 2^(al[i] - mx)  (== exp(alpha_i - mx) in nat units)
    float mx = fmaxf(fmaxf(al0, al1), fmaxf(al3, al4));
    float e0 = exp2r(al0 - mx);
    float e1 = exp2r(al1 - mx);
    float e3 = exp2r(al3 - mx);
    float e4 = exp2r(al4 - mx);

    // B operand (4x16 f32): VGPR0 = rows k0 (lanes 0-15) / k2 (lanes 16-31),
    // VGPR1 = rows k1 / k3.  All lanes hold valid e's -> pure select.
    v2f Bop;
    Bop.x = (lane < 16) ? e0 : e3;
    Bop.y = (lane < 16) ? e1 : e4;

    v8f c = {};
    v8f d = __builtin_amdgcn_wmma_f32_16x16x4_f32(false, Aop, false, Bop,
                                                  (short)0, c, false, false);
    // D: lane m and lane 16+m both hold s[jj] for batch m in d[jj].
    const float* et = ep + (size_t)t * LBL;
    if (t + 24 < S) __builtin_prefetch(et + 24 * LBL, 0, 3);  // global_prefetch_b8

    al0 = mx + log2r(d[0]) + et[0] * LOG2E;
    al1 = mx + log2r(d[1]) + et[1] * LOG2E;
    al3 = mx + log2r(d[3]) + et[3] * LOG2E;   // START dest: s==0 -> -inf (exact)
    al4 = mx + log2r(d[4]) + et[4] * LOG2E;

    if (t == len - 1) RECORD();
  }

  float contrib = valid ? fin * LN2 : 0.f;    // back to natural log
#pragma unroll
  for (int off = 16; off >= 1; off >>= 1)
    contrib += __shfl_xor(contrib, off, 32);
  if (lane == 0) partials[wave] = contrib;
#undef RECORD
}

// ---------------------------------------------------------------------------
// Gold-path ("labeled") score: pure gather-reduce over (b, t).
// ---------------------------------------------------------------------------
__global__ void __launch_bounds__(256)
crf_labeled(const float* __restrict__ lstm,
            const float* __restrict__ trans,
            const int*   __restrict__ lens,
            const int*   __restrict__ tags,
            float*       __restrict__ partials,
            int B, int S)
{
  __shared__ float red[8];
  const int tid    = blockIdx.x * blockDim.x + threadIdx.x;
  const int stride = gridDim.x * blockDim.x;
  const int total  = B * S;

  float acc = 0.f;
  for (int i = tid; i < total; i += stride) {
    int b = i / S, t = i - b * S;
    int len = lens[b];
    if (t < len) {
      int tg  = tags[(size_t)b * S + t];
      float v = lstm[((size_t)b * S + t) * LBL + tg];
      int prev = (t == 0) ? TSTART : tags[(size_t)b * S + t - 1];
      v += trans[prev * LBL + tg];
      if (t == len - 1) v += trans[tg * LBL + TEND];
      acc += v;
    }
  }
  // deterministic in-block reduction
#pragma unroll
  for (int off = 16; off >= 1; off >>= 1)
    acc += __shfl_xor(acc, off, 32);
  int lane = threadIdx.x & 31, w = threadIdx.x >> 5;
  if (lane == 0) red[w] = acc;
  __syncthreads();
  if (threadIdx.x < 32) {
    int nw = blockDim.x >> 5;
    float v = (threadIdx.x < nw) ? red[threadIdx.x] : 0.f;
#pragma unroll
    for (int off = 16; off >= 1; off >>= 1)
      v += __shfl_xor(v, off, 32);
    if (threadIdx.x == 0) partials[blockIdx.x] = v;
  }
}

// ---------------------------------------------------------------------------
// Deterministic finalize: fixed-order serial sums of the partials.
// ---------------------------------------------------------------------------
__global__ void finalize_k(const float* __restrict__ pf, int nf,
                           const float* __restrict__ pl, int nl,
                           float* __restrict__ out)
{
  if (threadIdx.x == 0 && blockIdx.x == 0) {
    float a = 0.f;
    for (int i = 0; i < nf; ++i) a += pf[i];
    float b = 0.f;
    for (int i = 0; i < nl; ++i) b += pl[i];
    out[0] = a;   // unlabeled (partition)
    out[1] = b;   // labeled (gold path)
  }
}

extern "C" void kernel_launch(void* const* d_in, const int* in_sizes, int n_in,
                              void* d_out, int out_size, void* d_ws, size_t ws_size,
                              hipStream_t stream)
{
  const float* lstm  = (const float*)d_in[0];
  const float* trans = (const float*)d_in[1];
  const int*   lens  = (const int*)  d_in[2];
  const int*   tags  = (const int*)  d_in[3];
  // d_in[4] (mask) ignored: recomputed as (t < len).

  const int B = in_sizes[2];
  const int S = in_sizes[3] / B;

  float* out = (float*)d_out;
  float* ws  = (float*)d_ws;

  const int nWaves  = (B + 15) / 16;      // forward partials
  const int lblocks = 256;                // labeled partials

  float* pf = ws;                         // [0, nWaves)
  float* pl = ws + nWaves;                // [nWaves, nWaves + lblocks)

  crf_forward_wmma<<<nWaves, 32, 0, stream>>>(lstm, trans, lens, pf, B, S);
  crf_labeled<<<lblocks, 256, 0, stream>>>(lstm, trans, lens, tags, pl, B, S);
  finalize_k<<<1, 32, 0, stream>>>(pf, nWaves, pl, lblocks, out);
}